// GELU280_22677427323087
// MI455X (gfx1250) — compile-verified
//
#include <hip/hip_runtime.h>
#include <hip/hip_bf16.h>

typedef __attribute__((ext_vector_type(2))) float v2f;
typedef __attribute__((ext_vector_type(8))) float v8f;

#define D_DIM    2048
#define RPB      8          // rows per block (amortize per-D vector loads)
#define NTHREADS 256
#define NWAVES   (NTHREADS / 32)

// Full 32-lane sum using V_WMMA_F32_16X16X4_F32.
// A (16x4 f32, 2 VGPRs): VGPR0 = per-lane value  -> A[m,0]=v[m], A[m,2]=v[m+16]
//                        VGPR1 = 0               -> A[m,1]=A[m,3]=0
// B (4x16) = all ones (layout independent)  =>  C[m,n] = v[m] + v[m+16]
// C layout: VGPR j holds row m=j (lanes 0-15) and m=j+8 (lanes 16-31),
// so sum_j C[j] = sum_{m=0..7} r[m] (lo half) / sum_{m=8..15} r[m] (hi half);
// one xor-16 shuffle completes the reduction (result in every lane).
__device__ __forceinline__ float wave_sum32(float v) {
    v2f a; a.x = v;    a.y = 0.0f;
    v2f b; b.x = 1.0f; b.y = 1.0f;
    v8f c = {};
    c = __builtin_amdgcn_wmma_f32_16x16x4_f32(false, a, false, b,
                                              (short)0, c, false, false);
    float t = c[0] + c[1] + c[2] + c[3] + c[4] + c[5] + c[6] + c[7];
    t += __shfl_xor(t, 16, 32);
    return t;
}

__device__ __forceinline__ float fast_tanhf(float a) {
    // tanh(a) = 1 - 2/(exp(2a)+1); saturates correctly for |a| large.
    float e = __expf(2.0f * a);
    return 1.0f - __fdividef(2.0f, e + 1.0f);
}

// ws layout (floats): [0,D) inv_std_in | [D,2D) inv_std_out |
//                     [2D,3D) ema_n    | [3D,4D) eval_n     | [4D,4D+8) scalars
__global__ __launch_bounds__(NTHREADS) void precompute_kernel(
    const float* __restrict__ ema_mean,    const float* __restrict__ ema_sq,
    const float* __restrict__ ema_out_mean,const float* __restrict__ ema_out_sq,
    const float* __restrict__ ema_out_dir, const float* __restrict__ eval_ema,
    const float* __restrict__ log_tau,     const float* __restrict__ log_beta_up,
    const float* __restrict__ log_beta_dn, const float* __restrict__ log_gamma,
    const float* __restrict__ log_beta_out,const float* __restrict__ log_gamma_out,
    const float* __restrict__ log_w_eval,  const float* __restrict__ logit_theta,
    float* __restrict__ ws)
{
    const int t = threadIdx.x;
    float sd = 0.0f, sev = 0.0f;
#pragma unroll
    for (int i = 0; i < D_DIM / NTHREADS; ++i) {
        int d = t + i * NTHREADS;
        float a = ema_out_dir[d]; sd  += a * a;
        float b = eval_ema[d];    sev += b * b;
    }
    sd  = wave_sum32(sd);
    sev = wave_sum32(sev);

    __shared__ float red[NWAVES][2];
    if ((t & 31) == 0) { red[t >> 5][0] = sd; red[t >> 5][1] = sev; }
    __syncthreads();
    float Sd = 0.0f, Se = 0.0f;
#pragma unroll
    for (int w = 0; w < NWAVES; ++w) { Sd += red[w][0]; Se += red[w][1]; }

    const float rnd = 1.0f / fmaxf(sqrtf(Sd), 1e-12f);
    const float rne = 1.0f / fmaxf(sqrtf(Se), 1e-12f);

    float* inv_std_in  = ws;
    float* inv_std_out = ws + D_DIM;
    float* ema_n       = ws + 2 * D_DIM;
    float* eval_n      = ws + 3 * D_DIM;
    float* scal        = ws + 4 * D_DIM;

#pragma unroll
    for (int i = 0; i < D_DIM / NTHREADS; ++i) {
        int d = t + i * NTHREADS;
        float m  = ema_mean[d];
        float v  = fmaxf(ema_sq[d] - m * m, 1e-4f);          // EPS_VAR
        inv_std_in[d] = 1.0f / (sqrtf(v) + 1e-5f);           // EPS
        float mo = ema_out_mean[d];
        float vo = fmaxf(ema_out_sq[d] - mo * mo, 1e-4f);
        inv_std_out[d] = 1.0f / (sqrtf(vo) + 1e-5f);
        ema_n[d]  = ema_out_dir[d] * rnd;
        eval_n[d] = eval_ema[d] * rne;
    }
    if (t == 0) {
        scal[0] = expf(log_tau[0]);                          // tau
        scal[1] = log1pf(expf(log_beta_up[0]));              // beta_up
        scal[2] = log1pf(expf(log_beta_dn[0]));              // beta_dn
        scal[3] = log1pf(expf(log_gamma[0]));                // gamma
        scal[4] = log1pf(expf(log_beta_out[0]));             // beta_out
        scal[5] = log1pf(expf(log_gamma_out[0]));            // gamma_out
        scal[6] = expf(log_w_eval[0]);                       // w_eval
        scal[7] = 1.0f / (1.0f + expf(-logit_theta[0]));     // theta
    }
}

#define LOAD8(dst, ptr)                                                        \
    { float4 _a = *(const float4*)((ptr) + d0);                                \
      float4 _b = *(const float4*)((ptr) + d1);                                \
      dst[0]=_a.x; dst[1]=_a.y; dst[2]=_a.z; dst[3]=_a.w;                      \
      dst[4]=_b.x; dst[5]=_b.y; dst[6]=_b.z; dst[7]=_b.w; }

__global__ __launch_bounds__(NTHREADS) void gated_gelu_kernel(
    const float* __restrict__ x,
    const float* __restrict__ ema_mean,
    const float* __restrict__ ema_out_mean,
    const float* __restrict__ ws,
    float* __restrict__ out, int rows)
{
    const int t  = threadIdx.x;
    const int d0 = 4 * t;            // float4 in first half-row
    const int d1 = 1024 + 4 * t;     // float4 in second half-row

    const float* inv_std_in  = ws;
    const float* inv_std_out = ws + D_DIM;
    const float* ema_n       = ws + 2 * D_DIM;
    const float* eval_n      = ws + 3 * D_DIM;
    const float* scal        = ws + 4 * D_DIM;

    const float tau      = scal[0], beta_up   = scal[1];
    const float beta_dn  = scal[2], gamma     = scal[3];
    const float beta_out = scal[4], gamma_out = scal[5];
    const float w_eval   = scal[6], theta     = scal[7];

    // Hoist all per-D vectors into registers for the whole row loop.
    float miv[8], iiv[8], mov[8], iov[8], env[8], evv[8];
    LOAD8(miv, ema_mean);
    LOAD8(iiv, inv_std_in);
    LOAD8(mov, ema_out_mean);
    LOAD8(iov, inv_std_out);
    LOAD8(env, ema_n);
    LOAD8(evv, eval_n);

    __shared__ float red[NWAVES][3];
    const float C0 = 0.7978845608028654f;  // sqrt(2/pi)

    for (int r = 0; r < RPB; ++r) {
        const long long row = (long long)blockIdx.x * RPB + r;
        if (row >= rows) break;                       // block-uniform
        const long long base = row * (long long)D_DIM;

        float xv[8];
        LOAD8(xv, x + base);
        if (row + 1 < rows) {                         // block-uniform
            __builtin_prefetch(x + base + D_DIM + d0, 0, 3);
            __builtin_prefetch(x + base + D_DIM + d1, 0, 3);
        }

        float prod[8];
        float s2 = 0.0f, sc = 0.0f, sev = 0.0f;
#pragma unroll
        for (int j = 0; j < 8; ++j) {
            float xj = xv[j];
            float o  = 0.5f * xj *
                       (1.0f + fast_tanhf(C0 * (xj + 0.044715f * xj * xj * xj)));
            float zi = (xj - miv[j]) * iiv[j];
            float ti = fast_tanhf(gamma * zi);
            float gin = 1.0f + beta_up * fmaxf(ti, 0.0f)
                             - beta_dn * fmaxf(-ti, 0.0f);
            gin = fminf(fmaxf(gin, 0.05f), 8.0f);
            float zo   = (o - mov[j]) * iov[j];
            float gout = 1.0f + beta_out * fast_tanhf(gamma_out * zo);
            gout = fminf(fmaxf(gout, 0.1f), 5.0f);
            prod[j] = o * gin * gout;
            s2  += o * o;
            sc  += o * env[j];
            sev += o * evv[j];
        }

        // Cross-lane reductions on the matrix pipe (v_wmma_f32_16x16x4_f32).
        s2  = wave_sum32(s2);
        sc  = wave_sum32(sc);
        sev = wave_sum32(sev);
        if ((t & 31) == 0) {
            int w = t >> 5;
            red[w][0] = s2; red[w][1] = sc; red[w][2] = sev;
        }
        __syncthreads();
        float S2 = 0.0f, SC = 0.0f, SE = 0.0f;
#pragma unroll
        for (int w = 0; w < NWAVES; ++w) {
            S2 += red[w][0]; SC += red[w][1]; SE += red[w][2];
        }
        __syncthreads();   // protect LDS reuse next iteration

        float rn = 1.0f / fmaxf(sqrtf(S2), 1e-12f);   // NEPS
        float cs = fminf(fmaxf(SC * rn, -1.0f), 1.0f);
        float ce = fminf(fmaxf(SE * rn, -1.0f), 1.0f);
        float g  = __expf(-tau * cs - w_eval * fmaxf(ce - theta, 0.0f));

        float4 oa, ob;
        oa.x = prod[0] * g; oa.y = prod[1] * g;
        oa.z = prod[2] * g; oa.w = prod[3] * g;
        ob.x = prod[4] * g; ob.y = prod[5] * g;
        ob.z = prod[6] * g; ob.w = prod[7] * g;
        *(float4*)(out + base + d0) = oa;
        *(float4*)(out + base + d1) = ob;
    }
}

extern "C" void kernel_launch(void* const* d_in, const int* in_sizes, int n_in,
                              void* d_out, int out_size, void* d_ws, size_t ws_size,
                              hipStream_t stream) {
    const float* x            = (const float*)d_in[0];
    const float* ema_mean     = (const float*)d_in[1];
    const float* ema_sq       = (const float*)d_in[2];
    const float* ema_out_mean = (const float*)d_in[3];
    const float* ema_out_sq   = (const float*)d_in[4];
    const float* ema_out_dir  = (const float*)d_in[5];
    const float* eval_ema     = (const float*)d_in[6];
    const float* log_tau      = (const float*)d_in[7];
    const float* log_beta_up  = (const float*)d_in[8];
    const float* log_beta_dn  = (const float*)d_in[9];
    const float* log_gamma    = (const float*)d_in[10];
    const float* log_beta_out = (const float*)d_in[11];
    const float* log_gamma_out= (const float*)d_in[12];
    const float* log_w_eval   = (const float*)d_in[13];
    const float* logit_theta  = (const float*)d_in[14];

    float* ws  = (float*)d_ws;
    float* out = (float*)d_out;
    const int rows = in_sizes[0] / D_DIM;   // B*T = 16384

    precompute_kernel<<<1, NTHREADS, 0, stream>>>(
        ema_mean, ema_sq, ema_out_mean, ema_out_sq, ema_out_dir, eval_ema,
        log_tau, log_beta_up, log_beta_dn, log_gamma,
        log_beta_out, log_gamma_out, log_w_eval, logit_theta, ws);

    const int blocks = (rows + RPB - 1) / RPB;
    gated_gelu_kernel<<<blocks, NTHREADS, 0, stream>>>(
        x, ema_mean, ema_out_mean, ws, out, rows);
}